// GCN_Guard_33603824124476
// MI455X (gfx1250) — compile-verified
//
#include <hip/hip_runtime.h>
#include <math.h>

#define NN 100000      // nodes
#define NE 1600000     // edges
#define DIN 128
#define DH  128
#define DOUT 64
#define SIM_THRESH 0.1f

typedef __attribute__((ext_vector_type(2))) float v2f;
typedef __attribute__((ext_vector_type(8))) float v8f;
typedef __attribute__((ext_vector_type(4))) int   v4i;

// ---- CDNA5 async global->LDS staging (guarded: falls back to plain copies) ----
#if defined(__HIP_DEVICE_COMPILE__) && defined(__gfx1250__) && \
    __has_builtin(__builtin_amdgcn_global_load_async_to_lds_b128)
#define ASYNC_LDS 1
#else
#define ASYNC_LDS 0
#endif

typedef __attribute__((address_space(1))) v4i GV4i;   // global int4
typedef __attribute__((address_space(3))) v4i LV4i;   // LDS int4

// ---------------- wave32 reductions ----------------
static __device__ __forceinline__ float waveSum(float v) {
    v += __shfl_xor(v, 1);
    v += __shfl_xor(v, 2);
    v += __shfl_xor(v, 4);
    v += __shfl_xor(v, 8);
    v += __shfl_xor(v, 16);
    return v;
}
static __device__ __forceinline__ float waveMax(float v) {
    v = fmaxf(v, __shfl_xor(v, 1));
    v = fmaxf(v, __shfl_xor(v, 2));
    v = fmaxf(v, __shfl_xor(v, 4));
    v = fmaxf(v, __shfl_xor(v, 8));
    v = fmaxf(v, __shfl_xor(v, 16));
    return v;
}

// ---------------- per-node inverse L2 norm (D==128), one wave per node ----------------
__global__ void k_node_invnorm(const float* __restrict__ X, float* __restrict__ invn) {
    int node = blockIdx.x * (blockDim.x >> 5) + (threadIdx.x >> 5);
    int lane = threadIdx.x & 31;
    if (node >= NN) return;
    const float4 v = *(const float4*)(X + (size_t)node * 128 + lane * 4);
    float s = v.x * v.x + v.y * v.y + v.z * v.z + v.w * v.w;
    s = waveSum(s);
    if (lane == 0) invn[node] = 1.0f / fmaxf(sqrtf(s), 1e-12f);
}

// ---------------- edge cosine similarity (D==128), one wave per edge ----------------
// validw == nullptr => all edges valid; else edge valid iff validw[e] > 0
__global__ void k_edge_sim(const float* __restrict__ X, const float* __restrict__ invn,
                           const int* __restrict__ row, const int* __restrict__ col,
                           const float* __restrict__ validw,
                           float* __restrict__ sim, float* __restrict__ rowsum) {
    int e = blockIdx.x * (blockDim.x >> 5) + (threadIdx.x >> 5);
    int lane = threadIdx.x & 31;
    if (e >= NE) return;
    int r = row[e], c = col[e];
    float s = 0.f;
    bool valid = (validw == nullptr) || (validw[e] > 0.f);
    if (valid) {
        const float4 a = *(const float4*)(X + (size_t)r * 128 + lane * 4);
        const float4 b = *(const float4*)(X + (size_t)c * 128 + lane * 4);
        s = a.x * b.x + a.y * b.y + a.z * b.z + a.w * b.w;
        s = waveSum(s);
        s *= invn[r] * invn[c];
        if (s < SIM_THRESH) s = 0.f;
    }
    if (lane == 0) {
        sim[e] = s;
        if (s > 0.f) atomicAdd(&rowsum[r], s);
    }
}

// ---- att = sim/rowsum; w = exp(att) on nz; deg += nz; wsum += w (fused segment sums) ----
__global__ void k_edge_att(const int* __restrict__ row, const float* __restrict__ sim,
                           const float* __restrict__ rowsum,
                           float* __restrict__ w, float* __restrict__ deg,
                           float* __restrict__ wsum) {
    int e = blockIdx.x * blockDim.x + threadIdx.x;
    if (e >= NE) return;
    int r = row[e];
    float s = sim[e];
    float rs = rowsum[r];
    float a = (rs > 0.f) ? s / rs : 0.f;
    if (a > 0.f) {
        float we = expf(a);
        w[e] = we;
        atomicAdd(&deg[r], 1.0f);
        atomicAdd(&wsum[r], we);
    } else {
        w[e] = 0.f;
    }
}

// ---- w_self = exp(1/(deg+1)); deg_w = wsum + w_self; dinv = rsqrt(deg_w) ----
__global__ void k_node_self(const float* __restrict__ deg, const float* __restrict__ wsum,
                            float* __restrict__ wself, float* __restrict__ dinv) {
    int n = blockIdx.x * blockDim.x + threadIdx.x;
    if (n >= NN) return;
    float ws = expf(1.0f / (deg[n] + 1.0f));
    float d = wsum[n] + ws;
    wself[n] = ws;
    dinv[n] = d > 0.f ? rsqrtf(d) : 0.f;
}

// ---------------- fp32 WMMA GEMM: C[M,Ncols] = A[M,K] * B[K,Ncols] ----------------
// 256-thread block (8 waves). Whole B panel (K x Ncols) staged in LDS via
// async global->LDS (CDNA5, ASYNCcnt) with a float4 fallback. Each wave owns one
// 16-wide column tile and up to RT=4 row tiles. The tile body is templated on
// the row-tile count so the hot loop is straight-line (no EXEC manipulation
// around WMMA, which requires EXEC all-ones).
// V_WMMA_F32_16X16X4_F32 layouts (ISA 7.12.2):
//   A frag (16x4): lane l -> m = l&15, koff = 2*(l>>4); vgpr j = A[m][k+koff+j]
//   B frag (4x16): lane l -> n = l&15, koff = 2*(l>>4); vgpr j = B[k+koff+j][n]
//   C/D (16x16):   vgpr r, lane l -> C[r + 8*(l>>4)][l&15]
#define GEMM_BLOCK 256
#define RT 4

template <int NRT>
static __device__ __forceinline__ void gemm_tiles(
    const float* __restrict__ A, const float* __restrict__ Bs,
    float* __restrict__ C, int K, int Ncols,
    int tmBase, int cn, int half, int idx)
{
    const float* arow[NRT];
#pragma unroll
    for (int r = 0; r < NRT; ++r)
        arow[r] = A + (size_t)(((tmBase + r) << 4) + idx) * K;

    v8f acc[NRT] = {};
#pragma unroll 4
    for (int k = 0; k < K; k += 4) {
        const int ka = k + (half << 1);
        v2f b;
        b.x = Bs[ka * Ncols + cn];
        b.y = Bs[(ka + 1) * Ncols + cn];
#pragma unroll
        for (int r = 0; r < NRT; ++r) {
            v2f a;
            a.x = arow[r][ka];
            a.y = arow[r][ka + 1];
            acc[r] = __builtin_amdgcn_wmma_f32_16x16x4_f32(
                false, a, false, b, (short)0, acc[r], false, false);
        }
    }

#pragma unroll
    for (int r = 0; r < NRT; ++r) {
        float* crow = C + (size_t)(((tmBase + r) << 4) + (half << 3)) * Ncols + cn;
#pragma unroll
        for (int j = 0; j < 8; ++j)
            crow[(size_t)j * Ncols] = acc[r][j];
    }
}

__global__ __launch_bounds__(GEMM_BLOCK)
void k_gemm_wmma_f32(const float* __restrict__ A, const float* __restrict__ B,
                     float* __restrict__ C, int M, int K, int Ncols) {
    extern __shared__ float Bs[];
    const int tid = threadIdx.x;

    // ---- stage B (K*Ncols floats) into LDS ----
    const int nChunks = (K * Ncols) >> 2;   // 16-byte chunks
    for (int i = tid; i < nChunks; i += GEMM_BLOCK) {
#if ASYNC_LDS
        __builtin_amdgcn_global_load_async_to_lds_b128(
            (GV4i*)(unsigned long long)(size_t)(B + (i << 2)),
            (LV4i*)(unsigned)(size_t)(Bs + (i << 2)),
            0, 0);
#else
        ((float4*)Bs)[i] = ((const float4*)B)[i];
#endif
    }
#if ASYNC_LDS
#if __has_builtin(__builtin_amdgcn_s_wait_asynccnt)
    __builtin_amdgcn_s_wait_asynccnt(0);
#else
    asm volatile("s_wait_asynccnt 0" ::: "memory");
#endif
#endif
    __syncthreads();

    const int wave = tid >> 5;
    const int lane = tid & 31;
    const int half = lane >> 4;
    const int idx  = lane & 15;

    const int NT = Ncols >> 4;                 // column tiles (8 for 128, 4 for 64)
    const int tn = wave % NT;                  // this wave's column tile
    const int rg = wave / NT;                  // row group within block
    const int rowTilesPerBlock = (8 / NT) * RT;
    const int numRowTiles = M >> 4;
    const int tmBase = blockIdx.x * rowTilesPerBlock + rg * RT;

    int nrt = numRowTiles - tmBase;            // valid row tiles for this wave
    if (nrt > RT) nrt = RT;
    if (nrt <= 0) return;                      // wave-uniform exit (after barrier)

    const int cn = (tn << 4) + idx;

    // uniform dispatch: each body is straight-line (EXEC untouched around WMMA)
    switch (nrt) {
    case 4: gemm_tiles<4>(A, Bs, C, K, Ncols, tmBase, cn, half, idx); break;
    case 3: gemm_tiles<3>(A, Bs, C, K, Ncols, tmBase, cn, half, idx); break;
    case 2: gemm_tiles<2>(A, Bs, C, K, Ncols, tmBase, cn, half, idx); break;
    default: gemm_tiles<1>(A, Bs, C, K, Ncols, tmBase, cn, half, idx); break;
    }
}

// ---------------- agg[row] += dinv[row]*w*dinv[col] * H[col], one wave per edge ----------------
__global__ void k_edge_aggregate(const int* __restrict__ row, const int* __restrict__ col,
                                 const float* __restrict__ w, const float* __restrict__ dinv,
                                 const float* __restrict__ H, float* __restrict__ AGG, int D) {
    int e = blockIdx.x * (blockDim.x >> 5) + (threadIdx.x >> 5);
    int lane = threadIdx.x & 31;
    if (e >= NE) return;
    float we = w[e];
    if (we <= 0.f) return;
    int r = row[e], c = col[e];
    float coef = dinv[r] * we * dinv[c];
    if (coef == 0.f) return;
    const float* hc = H + (size_t)c * D;
    float* ar = AGG + (size_t)r * D;
    for (int d = lane; d < D; d += 32)
        atomicAdd(&ar[d], coef * hc[d]);
}

// ---------------- layer-1 finalize: h = relu(agg + wself*dinv^2*hlin + b), in place ----------------
__global__ void k_finalize_relu(const float* __restrict__ hlin, float* __restrict__ agg,
                                const float* __restrict__ wself, const float* __restrict__ dinv,
                                const float* __restrict__ bias, int D) {
    long long i = (long long)blockIdx.x * blockDim.x + threadIdx.x;
    if (i >= (long long)NN * D) return;
    int n = (int)(i / D);
    int d = (int)(i - (long long)n * D);
    float di = dinv[n];
    float v = agg[i] + wself[n] * di * di * hlin[i] + bias[d];
    agg[i] = fmaxf(v, 0.f);
}

// ---------------- layer-2 finalize + log_softmax over 64 cols, one wave per node ----------------
__global__ void k_finalize_logsoftmax(const float* __restrict__ hlin2, const float* __restrict__ agg2,
                                      const float* __restrict__ wself, const float* __restrict__ dinv,
                                      const float* __restrict__ b2, float* __restrict__ out) {
    int node = blockIdx.x * (blockDim.x >> 5) + (threadIdx.x >> 5);
    int lane = threadIdx.x & 31;
    if (node >= NN) return;
    float di = dinv[node];
    float c = wself[node] * di * di;
    const float* hl = hlin2 + (size_t)node * 64;
    const float* ag = agg2 + (size_t)node * 64;
    float v0 = ag[lane]      + c * hl[lane]      + b2[lane];
    float v1 = ag[lane + 32] + c * hl[lane + 32] + b2[lane + 32];
    float m = waveMax(fmaxf(v0, v1));
    float s = waveSum(expf(v0 - m) + expf(v1 - m));
    float lse = m + logf(s);
    float* o = out + (size_t)node * 64;
    o[lane]      = v0 - lse;
    o[lane + 32] = v1 - lse;
}

extern "C" void kernel_launch(void* const* d_in, const int* in_sizes, int n_in,
                              void* d_out, int out_size, void* d_ws, size_t ws_size,
                              hipStream_t stream) {
    const float* x  = (const float*)d_in[0];
    const int*   ei = (const int*)d_in[1];
    const float* W1 = (const float*)d_in[2];
    const float* b1 = (const float*)d_in[3];
    const float* W2 = (const float*)d_in[4];
    const float* b2 = (const float*)d_in[5];
    float* out = (float*)d_out;

    const int* row = ei;          // edge_index[0]
    const int* col = ei + NE;     // edge_index[1]

    // ---- workspace layout (floats); rowsum/deg/wsum contiguous for one memset ----
    float* ws = (float*)d_ws;
    size_t off = 0;
    float* invn   = ws + off; off += NN;
    float* rowsum = ws + off; off += NN;
    float* deg    = ws + off; off += NN;
    float* wsum   = ws + off; off += NN;
    float* wself  = ws + off; off += NN;
    float* dinv   = ws + off; off += NN;
    float* att    = ws + off; off += NE;                // sim buffer (both passes)
    float* w1e    = ws + off; off += NE;                // pass-1 weights (nz1: w1e > 0)
    float* w2e    = ws + off; off += NE;
    float* H1     = ws + off; off += (size_t)NN * 128;  // hlin1; later hlin2 + agg2
    float* AGG    = ws + off; off += (size_t)NN * 128;  // agg1 -> h (in place)
    float* hlin2  = H1;
    float* agg2   = H1 + (size_t)NN * 64;

    const int BT = 256;
    dim3 bThread(BT);
    dim3 gEdgeT((NE + BT - 1) / BT);
    dim3 gNodeT((NN + BT - 1) / BT);
    dim3 gEdgeW((NE + 7) / 8);        // 8 waves per 256-thread block
    dim3 gNodeW((NN + 7) / 8);

    const int numRowTiles = NN / 16;  // 6250

    // =============== pass 1: attention on raw features ===============
    (void)hipMemsetAsync(rowsum, 0, 3 * NN * sizeof(float), stream);   // rowsum, deg, wsum
    k_node_invnorm<<<gNodeW, bThread, 0, stream>>>(x, invn);
    k_edge_sim<<<gEdgeW, bThread, 0, stream>>>(x, invn, row, col, nullptr, att, rowsum);
    k_edge_att<<<gEdgeT, bThread, 0, stream>>>(row, att, rowsum, w1e, deg, wsum);
    k_node_self<<<gNodeT, bThread, 0, stream>>>(deg, wsum, wself, dinv);

    // hlin1 = x @ W1  (fp32 WMMA, B panel in LDS)
    {
        int rtpb = (8 / (DH / 16)) * RT;                 // 4 row tiles / block
        dim3 grid((numRowTiles + rtpb - 1) / rtpb);
        size_t smem = (size_t)DIN * DH * sizeof(float);  // 64 KB
        k_gemm_wmma_f32<<<grid, dim3(GEMM_BLOCK), smem, stream>>>(x, W1, H1, NN, DIN, DH);
    }
    (void)hipMemsetAsync(AGG, 0, (size_t)NN * 128 * sizeof(float), stream);
    k_edge_aggregate<<<gEdgeW, bThread, 0, stream>>>(row, col, w1e, dinv, H1, AGG, DH);
    {
        long long total = (long long)NN * DH;
        dim3 g((unsigned)((total + BT - 1) / BT));
        k_finalize_relu<<<g, bThread, 0, stream>>>(H1, AGG, wself, dinv, b1, DH);
    }

    // =============== pass 2: attention on hidden features, valid = nz1 ===============
    (void)hipMemsetAsync(rowsum, 0, 3 * NN * sizeof(float), stream);
    k_node_invnorm<<<gNodeW, bThread, 0, stream>>>(AGG, invn);
    k_edge_sim<<<gEdgeW, bThread, 0, stream>>>(AGG, invn, row, col, w1e, att, rowsum);
    k_edge_att<<<gEdgeT, bThread, 0, stream>>>(row, att, rowsum, w2e, deg, wsum);
    k_node_self<<<gNodeT, bThread, 0, stream>>>(deg, wsum, wself, dinv);

    // hlin2 = h @ W2  (fp32 WMMA), into recycled H1 space
    {
        int rtpb = (8 / (DOUT / 16)) * RT;               // 8 row tiles / block
        dim3 grid((numRowTiles + rtpb - 1) / rtpb);
        size_t smem = (size_t)DH * DOUT * sizeof(float); // 32 KB
        k_gemm_wmma_f32<<<grid, dim3(GEMM_BLOCK), smem, stream>>>(AGG, W2, hlin2, NN, DH, DOUT);
    }
    (void)hipMemsetAsync(agg2, 0, (size_t)NN * 64 * sizeof(float), stream);
    k_edge_aggregate<<<gEdgeW, bThread, 0, stream>>>(row, col, w2e, dinv, hlin2, agg2, DOUT);
    k_finalize_logsoftmax<<<gNodeW, bThread, 0, stream>>>(hlin2, agg2, wself, dinv, b2, out);
}